// AttentionNet_17712445129186
// MI455X (gfx1250) — compile-verified
//
#include <hip/hip_runtime.h>
#include <math.h>

#define NFEAT 32
#define EMB   30
#define ATT   10
#define NPAIR 496          // 32*31/2
#define PPAD  512
#define KPAD  32
#define BLOCK 128
#define NWAVE (BLOCK / 32)

typedef __attribute__((ext_vector_type(2))) float v2f;
typedef __attribute__((ext_vector_type(8))) float v8f;

__global__ __launch_bounds__(BLOCK)
void afm_attention_kernel(const float* __restrict__ x,   // [B, 32, 30]
                          const float* __restrict__ aw,  // [30, 10]
                          const float* __restrict__ ab,  // [10]
                          const float* __restrict__ ah,  // [10]
                          const float* __restrict__ ap,  // [30, 1]
                          float* __restrict__ out)       // [B, 1]
{
    __shared__ float xs[NFEAT][KPAD];   // x[b] zero-padded to K=32
    __shared__ int   ii[PPAD];
    __shared__ int   jj[PPAD];
    __shared__ float sd[2][PPAD];       // [0] = score, [1] = prod . attention_p
    __shared__ float red[BLOCK];
    __shared__ float red2[BLOCK];

    const int tid  = threadIdx.x;
    const int lane = tid & 31;
    const int wave = tid >> 5;
    const int b    = blockIdx.x;

    // zero the two pad columns (disjoint from the loads below -> no race)
    if (tid < NFEAT * 2) xs[tid >> 1][EMB + (tid & 1)] = 0.0f;

    // stage x[b] into LDS (coalesced)
    const float* xb = x + (size_t)b * (NFEAT * EMB);
    for (int t = tid; t < NFEAT * EMB; t += BLOCK)
        xs[t / EMB][t % EMB] = xb[t];

    // upper-triangular pair tables (p -> (i,j), i<j); pad rows map to (0,0)
    for (int p = tid; p < PPAD; p += BLOCK) {
        int i = 0, j = 0;
        if (p < NPAIR) {
            int rem = p, cnt = NFEAT - 1;
            while (rem >= cnt) { rem -= cnt; --cnt; ++i; }
            j = i + 1 + rem;
        }
        ii[p] = i; jj[p] = j;
    }
    __syncthreads();

    // ---- WMMA: D = [W ; attention_p]^T x prod^T ----
    // M = attention dim (rows 0..9 = W^T, row 10 = attention_p, rest 0)
    // N = pair dim, K = embedding (padded 30 -> 32, 8 steps of K=4)
    const int ml    = lane & 15;          // A row / C-N pair-in-tile index
    const int hi    = lane >> 4;          // lane half
    const int khalf = hi * 2;             // K sub-offset per ISA fp32 layout

    // A fragments, invariant across tiles: v.x={K=k0}, v.y={K=k0+1} per half
    v2f af[8];
    #pragma unroll
    for (int ks = 0; ks < 8; ++ks) {
        const int k0 = ks * 4 + khalf;
        float a0 = 0.0f, a1 = 0.0f;
        if (k0 < EMB) {
            if (ml < ATT)       a0 = aw[k0 * ATT + ml];
            else if (ml == ATT) a0 = ap[k0];
        }
        if (k0 + 1 < EMB) {
            if (ml < ATT)       a1 = aw[(k0 + 1) * ATT + ml];
            else if (ml == ATT) a1 = ap[k0 + 1];
        }
        af[ks].x = a0; af[ks].y = a1;
    }
    // per-lane bias/h for the 8 attention rows this lane's C VGPRs hold:
    // VGPR r -> att = r + 8*hi ; zero beyond ATT (so the d-row adds nothing)
    float bn[8], hn[8];
    #pragma unroll
    for (int r = 0; r < 8; ++r) {
        const int a = r + 8 * hi;
        bn[r] = (a < ATT) ? ab[a] : 0.0f;
        hn[r] = (a < ATT) ? ah[a] : 0.0f;
    }

    // 32 N-tiles of 16 pairs; wave w handles tiles w, w+4, ...
    for (int nt = wave; nt < PPAD / 16; nt += NWAVE) {
        const int pr = nt * 16 + ml;
        const float* xi = xs[ii[pr]];
        const float* xj = xs[jj[pr]];
        v8f c = {};
        #pragma unroll
        for (int ks = 0; ks < 8; ++ks) {
            const int k0 = ks * 4 + khalf;
            v2f bfr;
            bfr.x = xi[k0]     * xj[k0];
            bfr.y = xi[k0 + 1] * xj[k0 + 1];
            c = __builtin_amdgcn_wmma_f32_16x16x4_f32(
                    false, af[ks], false, bfr, (short)0, c, false, false);
        }
        // score: private partial over this half's 8 att rows, one cross-half merge
        float s = 0.0f;
        #pragma unroll
        for (int r = 0; r < 8; ++r)
            s = fmaf(fmaxf(c[r] + bn[r], 0.0f), hn[r], s);
        s += __shfl_xor(s, 16);           // both halves now hold the full score
        // uniform store, no EXEC divergence:
        //   lanes 0-15  -> sd[0][p] = score
        //   lanes 16-31 -> sd[1][p] = c[2] = wx row M=10 = prod . attention_p
        sd[hi][nt * 16 + ml] = hi ? c[2] : s;
    }
    __syncthreads();

    // ---- softmax over 496 scores + weighted sum (single fused pass) ----
    float mx = -1e30f;
    for (int p = tid; p < NPAIR; p += BLOCK) mx = fmaxf(mx, sd[0][p]);
    red[tid] = mx; __syncthreads();
    for (int s = BLOCK / 2; s > 0; s >>= 1) {
        if (tid < s) red[tid] = fmaxf(red[tid], red[tid + s]);
        __syncthreads();
    }
    mx = red[0]; __syncthreads();

    float se = 0.0f, so = 0.0f;
    for (int p = tid; p < NPAIR; p += BLOCK) {
        const float e = __expf(sd[0][p] - mx);
        se += e;
        so = fmaf(e, sd[1][p], so);
    }
    red[tid] = se; red2[tid] = so; __syncthreads();
    for (int s = BLOCK / 2; s > 0; s >>= 1) {
        if (tid < s) { red[tid] += red[tid + s]; red2[tid] += red2[tid + s]; }
        __syncthreads();
    }
    if (tid == 0) out[b] = red2[0] / red[0];
}

extern "C" void kernel_launch(void* const* d_in, const int* in_sizes, int n_in,
                              void* d_out, int out_size, void* d_ws, size_t ws_size,
                              hipStream_t stream) {
    (void)n_in; (void)out_size; (void)d_ws; (void)ws_size;
    const float* x  = (const float*)d_in[0];
    const float* aw = (const float*)d_in[1];
    const float* ab = (const float*)d_in[2];
    const float* ah = (const float*)d_in[3];
    const float* ap = (const float*)d_in[4];
    float* out = (float*)d_out;

    const int nb = in_sizes[0] / (NFEAT * EMB);   // 8192 batches
    afm_attention_kernel<<<nb, BLOCK, 0, stream>>>(x, aw, ab, ah, ap, out);
}